// KmeansClustering_45835890983208
// MI455X (gfx1250) — compile-verified
//
#include <hip/hip_runtime.h>
#include <hip/hip_bf16.h>

// K-means single step, MI455X (gfx1250, wave32).
// dist GEMM fused with argmin via V_WMMA_F32_16X16X4_F32 (native fp32 WMMA),
// LDS tiles filled via GLOBAL_LOAD_ASYNC_TO_LDS_B128 when available.

#define N_PTS 16384
#define K_CLUS 1024
#define F_DIM 2048

typedef __attribute__((ext_vector_type(2))) float v2f;
typedef __attribute__((ext_vector_type(8))) float v8f;

#if defined(__has_builtin)
#if __has_builtin(__builtin_amdgcn_global_load_async_to_lds_b128) && \
    __has_builtin(__builtin_amdgcn_s_wait_asynccnt)
#define HAVE_ASYNC_LDS 1
// Exact parameter types per hipcc diagnostic: arg0 is
// "int __vector(4) __device__ *" i.e. address_space(1) pointer to a 16-byte
// int vector; arg1 is the LDS-side (address_space(3)) counterpart.
typedef __attribute__((__vector_size__(4 * sizeof(int)))) int vi4;
typedef __attribute__((address_space(1))) vi4* gvi4p;  // global (AS1)
typedef __attribute__((address_space(3))) vi4* svi4p;  // LDS (AS3)
#endif
#endif

// ---------------------------------------------------------------------------
// Kernel 1: c2[k] = sum_f C[k,f]^2           (grid = K, block = 256)
// ---------------------------------------------------------------------------
__global__ __launch_bounds__(256) void c2_kernel(const float* __restrict__ cent,
                                                 float* __restrict__ c2) {
  __shared__ float red[256];
  const int k = blockIdx.x, t = threadIdx.x;
  float s = 0.f;
  for (int f = t; f < F_DIM; f += 256) {
    float v = cent[(size_t)k * F_DIM + f];
    s += v * v;
  }
  red[t] = s; __syncthreads();
  for (int off = 128; off > 0; off >>= 1) {
    if (t < off) red[t] += red[t + off];
    __syncthreads();
  }
  if (t == 0) c2[k] = red[0];
}

// ---------------------------------------------------------------------------
// Kernel 2: csum[f] = sum_k C[k,f]           (grid = F/256, block = 256)
// ---------------------------------------------------------------------------
__global__ __launch_bounds__(256) void csum_kernel(const float* __restrict__ cent,
                                                   float* __restrict__ csum) {
  const int f = blockIdx.x * 256 + threadIdx.x;
  float s = 0.f;
  for (int k = 0; k < K_CLUS; ++k) s += cent[(size_t)k * F_DIM + f];
  csum[f] = s;
}

// ---------------------------------------------------------------------------
// Kernel 3: c2sum = sum_k c2[k]              (grid = 1, block = 256)
// ---------------------------------------------------------------------------
__global__ __launch_bounds__(256) void c2sum_kernel(const float* __restrict__ c2,
                                                    float* __restrict__ scal) {
  __shared__ float red[256];
  const int t = threadIdx.x;
  float s = 0.f;
  for (int k = t; k < K_CLUS; k += 256) s += c2[k];
  red[t] = s; __syncthreads();
  for (int off = 128; off > 0; off >>= 1) {
    if (t < off) red[t] += red[t + off];
    __syncthreads();
  }
  if (t == 0) scal[0] = red[0];  // c2sum
}

// ---------------------------------------------------------------------------
// Kernel 4: colsum[n] = K*x2[n] + c2sum - 2*dot(csum, x_n)
//           (grid = N, block = 256) -- single fused pass over x
// ---------------------------------------------------------------------------
__global__ __launch_bounds__(256) void colsum_kernel(const float* __restrict__ x,
                                                     const float* __restrict__ csum,
                                                     const float* __restrict__ scal,
                                                     float* __restrict__ colsum) {
  __shared__ float redA[256];
  __shared__ float redB[256];
  const int n = blockIdx.x, t = threadIdx.x;
  float sx = 0.f, sd = 0.f;
  for (int f = t; f < F_DIM; f += 256) {
    float v = x[(size_t)n * F_DIM + f];
    sx += v * v;
    sd += csum[f] * v;
  }
  redA[t] = sx; redB[t] = sd; __syncthreads();
  for (int off = 128; off > 0; off >>= 1) {
    if (t < off) { redA[t] += redA[t + off]; redB[t] += redB[t + off]; }
    __syncthreads();
  }
  if (t == 0)
    colsum[n] = (float)K_CLUS * redA[0] + scal[0] - 2.f * redB[0];
}

// ---------------------------------------------------------------------------
// Kernel 5 (main): fused GEMM + argmin via WMMA f32 16x16x4.
// Block = 256 threads (8 waves). Each wave owns a 16-point column tile; the
// block covers 128 points. K is processed in chunks of 128 (8 k-tiles of 16,
// v8f acc[8] per wave). F is staged through LDS in chunks of 64 floats; the
// X panel is reused across all 8 k-tiles per fill.
//
// WMMA f32 16x16x4 lane layout (wave32):
//   A (16x4): lanes 0-15 row m=lane hold f={0,1}; lanes 16-31 hold f={2,3}
//             -> one aligned 8-byte LDS load per lane (ds_load_b64).
//   B (4x16): mirrored; lane%16 = column n.
//   D (16x16): vgpr j, lanes 0-15 -> row m=j, lanes 16-31 -> row m=j+8,
//              column n = lane%16.
// ---------------------------------------------------------------------------
#define NB 128      // points per block
#define KB 128      // centroids per accumulation chunk
#define CF 64       // F chunk staged in LDS
#define LSTR 68     // LDS row stride (floats), 272B: 16B aligned, depadded

__global__ __launch_bounds__(256) void nearest_kernel(const float* __restrict__ x,
                                                      const float* __restrict__ cent,
                                                      const float* __restrict__ c2,
                                                      int* __restrict__ nearest) {
  __shared__ float Xs[NB * LSTR];
  __shared__ float Cs[KB * LSTR];

  const int tid  = threadIdx.x;
  const int wave = tid >> 5;       // 0..7
  const int lane = tid & 31;
  const int half = lane >> 4;      // 0 or 1
  const int l16  = lane & 15;
  const int n0   = blockIdx.x * NB;
  const int nw   = wave * 16;      // wave's 16-column slice

  float bestv = 3.402823466e+38f;
  int   besti = 0;

  for (int kb = 0; kb < K_CLUS; kb += KB) {
    v8f acc[8] = {};

    for (int fb = 0; fb < F_DIM; fb += CF) {
      __syncthreads();  // LDS consumed by previous iteration
      // Cooperative fill: 128 rows x 16 float4 slots per array.
#ifdef HAVE_ASYNC_LDS
      for (int s = tid; s < NB * (CF / 4); s += 256) {
        const int row = s >> 4;
        const int c4  = (s & 15) << 2;
        // L2 -> LDS, no VGPR staging; tracked by ASYNCcnt.
        __builtin_amdgcn_global_load_async_to_lds_b128(
            (gvi4p)(uintptr_t)(x + (size_t)(n0 + row) * F_DIM + fb + c4),
            (svi4p)(unsigned int)(uintptr_t)&Xs[row * LSTR + c4],
            0, 0);
        __builtin_amdgcn_global_load_async_to_lds_b128(
            (gvi4p)(uintptr_t)(cent + (size_t)(kb + row) * F_DIM + fb + c4),
            (svi4p)(unsigned int)(uintptr_t)&Cs[row * LSTR + c4],
            0, 0);
      }
      __builtin_amdgcn_s_wait_asynccnt(0);
#else
      for (int s = tid; s < NB * (CF / 4); s += 256) {
        const int row = s >> 4;
        const int c4  = (s & 15) << 2;
        float4 xv = *(const float4*)(x    + (size_t)(n0 + row) * F_DIM + fb + c4);
        float4 cv = *(const float4*)(cent + (size_t)(kb + row) * F_DIM + fb + c4);
        *(float4*)&Xs[row * LSTR + c4] = xv;
        *(float4*)&Cs[row * LSTR + c4] = cv;
      }
#endif
      __syncthreads();

      for (int fs = 0; fs < CF; fs += 4) {
        // B operand (x tile): one ds_load_b64, shared across all 8 k-tiles.
        const v2f b = *(const v2f*)&Xs[(nw + l16) * LSTR + fs + 2 * half];
#pragma unroll
        for (int kt = 0; kt < 8; ++kt) {
          const v2f a = *(const v2f*)&Cs[(kt * 16 + l16) * LSTR + fs + 2 * half];
          acc[kt] = __builtin_amdgcn_wmma_f32_16x16x4_f32(
              /*neg_a=*/false, a, /*neg_b=*/false, b,
              /*c_mod=*/(short)0, acc[kt],
              /*reuse_a=*/false, /*reuse_b=*/false);
        }
      }
    }

    // Argmin update: score = c2[k] - 2*dot (x2[n] is constant over k).
#pragma unroll
    for (int kt = 0; kt < 8; ++kt) {
#pragma unroll
      for (int j = 0; j < 8; ++j) {
        const int k = kb + kt * 16 + j + half * 8;
        const float score = c2[k] - 2.f * acc[kt][j];
        if (score < bestv) { bestv = score; besti = k; }
      }
    }
  }

  // Lanes L and L+16 hold the same column n: combine halves.
  const float ov = __shfl_xor(bestv, 16, 32);
  const int   oi = __shfl_xor(besti, 16, 32);
  if (ov < bestv || (ov == bestv && oi < besti)) { bestv = ov; besti = oi; }
  if (lane < 16) nearest[n0 + nw + l16] = besti;
}

// ---------------------------------------------------------------------------
// Kernel 6: counts[k] += 1 per assigned point; avgpartial += colsum[i]
//           (grid = N/256, block = 256)
// ---------------------------------------------------------------------------
__global__ __launch_bounds__(256) void counts_kernel(const int* __restrict__ nearest,
                                                     const float* __restrict__ colsum,
                                                     float* __restrict__ counts,
                                                     float* __restrict__ scal) {
  __shared__ float red[256];
  const int i = blockIdx.x * 256 + threadIdx.x;
  const int t = threadIdx.x;
  atomicAdd(&counts[nearest[i]], 1.0f);
  red[t] = colsum[i]; __syncthreads();
  for (int off = 128; off > 0; off >>= 1) {
    if (t < off) red[t] += red[t + off];
    __syncthreads();
  }
  if (t == 0) atomicAdd(&scal[3], red[0]);  // avgpartial
}

// ---------------------------------------------------------------------------
// Kernel 7: sums[nearest[i], :] += x[i, :]   (grid = N, block = 256)
// ---------------------------------------------------------------------------
__global__ __launch_bounds__(256) void sums_kernel(const float* __restrict__ x,
                                                   const int* __restrict__ nearest,
                                                   float* __restrict__ sums) {
  const int i = blockIdx.x;
  const int kk = nearest[i];
  for (int f = threadIdx.x; f < F_DIM; f += 256)
    atomicAdd(&sums[(size_t)kk * F_DIM + f], x[(size_t)i * F_DIM + f]);
}

// ---------------------------------------------------------------------------
// Kernel 8: delta[k] = ||c_k - new_k||^2 for nonempty k; accumulate
//           sum_delta (scal[1]) and nonempty count (scal[2]). (grid = K)
// ---------------------------------------------------------------------------
__global__ __launch_bounds__(256) void finalize_kernel(const float* __restrict__ cent,
                                                       const float* __restrict__ sums,
                                                       const float* __restrict__ counts,
                                                       float* __restrict__ scal) {
  __shared__ float red[256];
  const int k = blockIdx.x, t = threadIdx.x;
  const float cnt = counts[k];
  float s = 0.f;
  if (cnt > 0.f) {
    const float inv = 1.0f / cnt;
    for (int f = t; f < F_DIM; f += 256) {
      const float nc = sums[(size_t)k * F_DIM + f] * inv;
      const float d  = cent[(size_t)k * F_DIM + f] - nc;
      s += d * d;
    }
  }
  red[t] = s; __syncthreads();
  for (int off = 128; off > 0; off >>= 1) {
    if (t < off) red[t] += red[t + off];
    __syncthreads();
  }
  if (t == 0 && cnt > 0.f) {
    atomicAdd(&scal[1], red[0]);  // sum_delta
    atomicAdd(&scal[2], 1.0f);   // nonempty count
  }
}

// ---------------------------------------------------------------------------
// Kernel 9: write the two scalar outputs.
// ---------------------------------------------------------------------------
__global__ void writeout_kernel(const float* __restrict__ scal,
                                float* __restrict__ out) {
  const float dk = scal[2] > 0.f ? scal[2] : 1.f;
  out[0] = scal[1] / dk;                       // sum_delta / delta_k
  out[1] = scal[3] / (float)N_PTS;             // avg_dist
}

// ---------------------------------------------------------------------------
extern "C" void kernel_launch(void* const* d_in, const int* in_sizes, int n_in,
                              void* d_out, int out_size, void* d_ws, size_t ws_size,
                              hipStream_t stream) {
  const float* x    = (const float*)d_in[0];   // [N, F]
  const float* cent = (const float*)d_in[1];   // [K, F]
  float* out = (float*)d_out;                  // [2]

  // Workspace layout (floats). Total ~8.6 MB.
  float* ws      = (float*)d_ws;
  float* sums    = ws;                                   // K*F
  float* counts  = sums + (size_t)K_CLUS * F_DIM;        // K
  float* scal    = counts + K_CLUS;                      // 4: c2sum, sum_delta, nonempty, avgpartial
  float* c2      = scal + 4;                             // K
  float* csum    = c2 + K_CLUS;                          // F
  float* colsum  = csum + F_DIM;                         // N
  int*   nearest = (int*)(colsum + N_PTS);               // N

  // Zero the atomic accumulators (sums | counts | scal are contiguous).
  (void)hipMemsetAsync(ws, 0,
      ((size_t)K_CLUS * F_DIM + K_CLUS + 4) * sizeof(float), stream);

  c2_kernel      <<<K_CLUS,       256, 0, stream>>>(cent, c2);
  csum_kernel    <<<F_DIM / 256,  256, 0, stream>>>(cent, csum);
  c2sum_kernel   <<<1,            256, 0, stream>>>(c2, scal);
  colsum_kernel  <<<N_PTS,        256, 0, stream>>>(x, csum, scal, colsum);
  nearest_kernel <<<N_PTS / NB,   256, 0, stream>>>(x, cent, c2, nearest);
  counts_kernel  <<<N_PTS / 256,  256, 0, stream>>>(nearest, colsum, counts, scal);
  sums_kernel    <<<N_PTS,        256, 0, stream>>>(x, nearest, sums);
  finalize_kernel<<<K_CLUS,       256, 0, stream>>>(cent, sums, counts, scal);
  writeout_kernel<<<1,            1,   0, stream>>>(scal, out);
}